// RecurrentLinearAttention_17171279249476
// MI455X (gfx1250) — compile-verified
//
#include <hip/hip_runtime.h>

// RecurrentLinearAttention, single step, fused.
// Shapes: B=32, N=32, D=256, M=256. All f32.
// Memory-bound: Si traffic = 512 MiB r+w  =>  ~23 us at 23.3 TB/s.
// Strategy: 1 block per (b,n) token, 16 wave32 waves, each wave owns a
// 16-wide m-tile. Rank-1 update of Si streamed non-temporally; the updated
// values feed V_WMMA_F32_16X16X4_F32 directly (B operand layout) to form
// V = Q^T * Si_new on the matrix pipe.

typedef float v2f __attribute__((ext_vector_type(2)));
typedef float v8f __attribute__((ext_vector_type(8)));

#define DB 32
#define DN 32
#define DD 256
#define DM 256
#define EPS 1e-6f

__device__ __forceinline__ float elu_fm(float x) {
    // elu(x) + 1 == x+1 (x>0) else exp(x)
    return x > 0.0f ? x + 1.0f : __expf(x);
}

__global__ __launch_bounds__(512) void rla_step_kernel(
    const float* __restrict__ q_in,   // [B,N,D]
    const float* __restrict__ k_in,   // [B,N,D]
    const float* __restrict__ v_in,   // [B,N,M]
    const float* __restrict__ Si_in,  // [B,N,D,M]
    const float* __restrict__ Zi_in,  // [B,N,D]
    float* __restrict__ V_out,        // [B,N,M]
    float* __restrict__ Si_out,       // [B,N,D,M]
    float* __restrict__ Zi_out)       // [B,N,D]
{
    __shared__ float sQ[DD];
    __shared__ float sK[DD];
    __shared__ float red[DD + 1];   // red[DD] holds Z

    const int  t   = threadIdx.x;
    const long tok = (long)blockIdx.x;           // token index bn in [0,1024)
    const long vecBase = tok * DD;               // q/k/Zi/value offset
    const long matBase = tok * (long)(DD * DM);  // Si offset

    // ---- Prologue: feature maps, Zi_new, and Z = 1/(Q . Zi_new + eps) ----
    if (t < DD) {
        float Qf = elu_fm(q_in[vecBase + t]);
        float Kf = elu_fm(k_in[vecBase + t]);
        sQ[t] = Qf;
        sK[t] = Kf;
        float zn = Zi_in[vecBase + t] + Kf;
        Zi_out[vecBase + t] = zn;
        red[t] = Qf * zn;
    }
    __syncthreads();
    #pragma unroll
    for (int s = DD / 2; s > 0; s >>= 1) {
        if (t < s) red[t] += red[t + s];
        __syncthreads();
    }
    if (t == 0) red[DD] = 1.0f / (red[0] + EPS);
    __syncthreads();
    const float Z = red[DD];

    // ---- Main stream: rank-1 update of Si + WMMA reduction for V ----
    const int lane = t & 31;
    const int wave = t >> 5;          // 16 waves
    const int m0   = wave * 16;       // 16-column m-tile per wave
    const int half = lane >> 4;       // 0: lanes 0-15, 1: lanes 16-31
    const int col  = lane & 15;       // column within tile

    const float vcol = v_in[vecBase + m0 + col];  // value[m] constant per lane

    const float* SiP = Si_in  + matBase;
    float*       SoP = Si_out + matBase;

    v8f acc = {};  // all 16 rows of the accumulator will be identical

    for (int d = 0; d < DD; d += 4) {
        const int r0 = d + 2 * half;             // row for VGPR0 of A/B
        // Q/K pairs for rows r0, r0+1 (r0 even -> 8B aligned LDS reads)
        const float2 qq = *(const float2*)&sQ[r0];
        const float2 kk = *(const float2*)&sK[r0];

        const long off0 = (long)r0 * DM + m0 + col;
        // Stream Si non-temporally (512 MiB >> 192 MB L2)
        float s0 = __builtin_nontemporal_load(&SiP[off0]);
        float s1 = __builtin_nontemporal_load(&SiP[off0 + DM]);
        s0 = fmaf(kk.x, vcol, s0);               // Si_new = Si + K (x) value
        s1 = fmaf(kk.y, vcol, s1);
        __builtin_nontemporal_store(s0, &SoP[off0]);
        __builtin_nontemporal_store(s1, &SoP[off0 + DM]);

        // A (16x4 f32): every row = Q[d..d+3]  (lane half h: Q[d+2h],Q[d+2h+1])
        // B (4x16 f32): rows d..d+3 of updated Si for this m-tile
        v2f a = { qq.x, qq.y };
        v2f b = { s0,   s1   };
        acc = __builtin_amdgcn_wmma_f32_16x16x4_f32(
                  false, a, false, b, (short)0, acc, false, false);
    }

    // D layout: VGPR0 lanes 0-15 = (M=0, N=lane); all rows identical.
    if (lane < 16) {
        V_out[vecBase + m0 + col] = acc[0] * Z;
    }
}

extern "C" void kernel_launch(void* const* d_in, const int* in_sizes, int n_in,
                              void* d_out, int out_size, void* d_ws, size_t ws_size,
                              hipStream_t stream) {
    (void)in_sizes; (void)n_in; (void)d_ws; (void)ws_size; (void)out_size;

    const float* q  = (const float*)d_in[0];  // query [32,32,256]
    const float* k  = (const float*)d_in[1];  // key   [32,32,256]
    const float* v  = (const float*)d_in[2];  // value [32,32,256]
    const float* Si = (const float*)d_in[3];  // Si    [32,32,256,256]
    const float* Zi = (const float*)d_in[4];  // Zi    [32,32,256]

    const size_t nVec = (size_t)DB * DN * DD;       // 262,144
    const size_t nMat = (size_t)DB * DN * DD * DM;  // 67,108,864

    float* out    = (float*)d_out;
    float* V_out  = out;                 // [B,N,M]
    float* Si_out = out + nVec;          // [B,N,D,M]
    float* Zi_out = out + nVec + nMat;   // [B,N,D]

    dim3 grid(DB * DN);   // 1024 blocks, one per token
    dim3 block(512);      // 16 wave32 waves
    rla_step_kernel<<<grid, block, 0, stream>>>(q, k, v, Si, Zi,
                                                V_out, Si_out, Zi_out);
}